// PrototypeManager_72533407695471
// MI455X (gfx1250) — compile-verified
//
#include <hip/hip_runtime.h>

#define NCLASS   21
#define KPAD     32          // class rows padded to 2 WMMA M-tiles (rows 21..31 stay zero)
#define BIMG     8
#define CCH      256
#define SRC      64
#define SCELLS   (SRC * SRC) // 4096
#define EPSV     1e-6f

typedef __attribute__((ext_vector_type(2))) float v2f;
typedef __attribute__((ext_vector_type(8))) float v8f;

// ---------------------------------------------------------------------------
// Kernel 0: zero workspace (W + counts) and output accumulator
// ---------------------------------------------------------------------------
__global__ void proto_zero(float* __restrict__ ws, int nws,
                           float* __restrict__ out, int nout) {
    int stride = gridDim.x * blockDim.x;
    for (int j = blockIdx.x * blockDim.x + threadIdx.x; j < nws;  j += stride) ws[j]  = 0.f;
    for (int j = blockIdx.x * blockDim.x + threadIdx.x; j < nout; j += stride) out[j] = 0.f;
}

// ---------------------------------------------------------------------------
// Kernel 1: fold bilinear resize into per-class weight field
//   W[b,k,s] += bilinear tap weights of every output pixel with label k
//   counts[b,k] += #pixels with label k   (LDS histogram -> 1 atomic/class/WG)
// ---------------------------------------------------------------------------
__global__ void proto_scatter(const int* __restrict__ masks,
                              float* __restrict__ W,
                              float* __restrict__ counts) {
    __shared__ int hist[NCLASS];
    if (threadIdx.x < NCLASS) hist[threadIdx.x] = 0;
    __syncthreads();

    int gid = blockIdx.x * 256 + threadIdx.x;   // 0 .. 8*256*256-1
    int b   = gid >> 16;
    int p   = gid & 65535;
    int y   = p >> 8;
    int x   = p & 255;

    int lbl = masks[gid];
    if ((unsigned)lbl < (unsigned)NCLASS) {     // label==NCLASS (dump) & OOR drop out
        atomicAdd(&hist[lbl], 1);

        // half-pixel bilinear, 4x upsample, edge-clamped (== jax/torch bilinear)
        float sx = (x + 0.5f) * 0.25f - 0.5f;
        float sy = (y + 0.5f) * 0.25f - 0.5f;
        float fx0 = floorf(sx), fy0 = floorf(sy);
        float wx1 = sx - fx0,   wy1 = sy - fy0;
        float wx0 = 1.f - wx1,  wy0 = 1.f - wy1;
        int x0 = (int)fx0, y0 = (int)fy0;
        int x1 = min(x0 + 1, SRC - 1), y1 = min(y0 + 1, SRC - 1);
        x0 = max(x0, 0);  y0 = max(y0, 0);

        float* Wb = W + ((size_t)(b * KPAD + lbl)) * SCELLS;
        atomicAdd(Wb + y0 * SRC + x0, wy0 * wx0);
        atomicAdd(Wb + y0 * SRC + x1, wy0 * wx1);
        atomicAdd(Wb + y1 * SRC + x0, wy1 * wx0);
        atomicAdd(Wb + y1 * SRC + x1, wy1 * wx1);
    }
    __syncthreads();
    if (threadIdx.x < NCLASS) {
        int h = hist[threadIdx.x];
        if (h) atomicAdd(&counts[b * KPAD + threadIdx.x], (float)h);
    }
}

// ---------------------------------------------------------------------------
// Kernel 2: per-image GEMM  D[32,256] = W[32,4096] x feats^T[4096,256]
//   via V_WMMA_F32_16X16X4_F32, K split 4-ways across waves.
//   Epilogue: out[k,c] += D * 0.125 / (count+eps)  (atomic merge)
// 1024 waves = 8 imgs * 2 Mtiles * 16 Ntiles * 4 Ksegs; 256 wmma/wave.
// ---------------------------------------------------------------------------
__global__ void proto_gemm(const float* __restrict__ feats,
                           const float* __restrict__ W,
                           const float* __restrict__ counts,
                           float* __restrict__ out) {
    int wave  = blockIdx.x * 8 + (threadIdx.x >> 5);
    int lane  = threadIdx.x & 31;
    int kseg  = wave & 3;           // 1024 K-values per segment
    int ntile = (wave >> 2) & 15;   // 16 channels
    int mtile = (wave >> 6) & 1;    // class rows 0-15 / 16-31
    int b     = wave >> 7;

    int half = lane >> 4;           // K sub-lane split per ISA fragment layout
    int l15  = lane & 15;

    // A: lane holds W[b, mtile*16 + l15, k0 + 2*half + {0,1}]  (contiguous f32x2)
    const float* Arow = W     + ((size_t)(b * KPAD + mtile * 16 + l15)) * SCELLS
                              + kseg * 1024 + 2 * half;
    // B: lane holds feats[b, ntile*16 + l15, k0 + 2*half + {0,1}] (contiguous f32x2)
    const float* Brow = feats + ((size_t)b * CCH + ntile * 16 + l15) * SCELLS
                              + kseg * 1024 + 2 * half;

    v8f acc = {0.f, 0.f, 0.f, 0.f, 0.f, 0.f, 0.f, 0.f};

#pragma unroll 8
    for (int k0 = 0; k0 < 1024; k0 += 4) {
        v2f a  = *(const v2f*)(Arow + k0);
        v2f bm = *(const v2f*)(Brow + k0);
        acc = __builtin_amdgcn_wmma_f32_16x16x4_f32(
                  /*neg_a=*/false, a, /*neg_b=*/false, bm,
                  /*c_mod=*/(short)0, acc, /*reuse_a=*/false, /*reuse_b=*/false);
    }

    // C/D layout: vgpr i, lane -> row m = i + 8*half, col n = l15
    int cbase = ntile * 16;
#pragma unroll
    for (int i = 0; i < 8; ++i) {
        int m = mtile * 16 + i + 8 * half;
        if (m < NCLASS) {
            float cnt   = counts[b * KPAD + m];
            float scale = 0.125f / (cnt + EPSV);     // mean over 8 imgs / (count+eps)
            atomicAdd(&out[m * CCH + cbase + l15], acc[i] * scale);
        }
    }
}

// ---------------------------------------------------------------------------
extern "C" void kernel_launch(void* const* d_in, const int* in_sizes, int n_in,
                              void* d_out, int out_size, void* d_ws, size_t ws_size,
                              hipStream_t stream) {
    const float* feats = (const float*)d_in[0];   // [8,256,64,64] f32
    const int*   masks = (const int*)d_in[1];     // [8,256,256]   i32
    float*       out   = (float*)d_out;           // [21,256]      f32

    float* W      = (float*)d_ws;                 // [8,32,4096] f32 = 4 MB
    const int nW  = BIMG * KPAD * SCELLS;
    float* counts = W + nW;                       // [8,32] f32
    const int nws = nW + BIMG * KPAD;

    proto_zero   <<<1024, 256, 0, stream>>>(W, nws, out, out_size);
    proto_scatter<<<BIMG * 256, 256, 0, stream>>>(masks, W, counts);
    proto_gemm   <<<128, 256, 0, stream>>>(feats, W, counts, out);
}